// Ranking_Module_36567351558724
// MI455X (gfx1250) — compile-verified
//
#include <hip/hip_runtime.h>
#include <hip/hip_bf16.h>

// ---------------------------------------------------------------------------
// CDNA5 / gfx1250 implementation of the GNN ranking module.
// f32 matrix work mapped onto V_WMMA_F32_16X16X4_F32 (one wave32 per 16x16
// output tile, K in chunks of 4). Per-edge NNConv fused (theta never
// materialized), weights staged in LDS. Scatter via global_atomic_add_f32.
// ---------------------------------------------------------------------------

#define NN   10000
#define EE   40000
#define FIN  8
#define HH   64

typedef __attribute__((ext_vector_type(2))) float v2f;
typedef __attribute__((ext_vector_type(8))) float v8f;

__device__ __forceinline__ v8f wmma_f32_4(v2f a, v2f b, v8f c) {
    // D = A(16x4) * B(4x16) + C(16x16), all f32, wave32
    return __builtin_amdgcn_wmma_f32_16x16x4_f32(
        /*neg_a=*/false, a, /*neg_b=*/false, b,
        /*c_mod=*/(short)0, c, /*reuse_a=*/false, /*reuse_b=*/false);
}

// Branch-free K-segment accumulation: acc += A[m, 0:kcnt] * B[kbase:kbase+kcnt, n]
// A-matrix layout (32b, 16x4): lanes 0-15 hold K=k0,k0+1; lanes 16-31 K=k0+2,k0+3.
// B layout (4x16): VGPR0 = rows K=k0 (lanes 0-15) / K=k0+2 (lanes 16-31), VGPR1 = +1.
__device__ __forceinline__ v8f gemm_seg(v8f acc,
                                        const float* __restrict__ A, int lda, int m,
                                        const float* __restrict__ B, int n,
                                        int kbase, int kcnt, int half) {
    for (int k0 = 0; k0 < kcnt; k0 += 4) {
        int ka = k0 + 2 * half;
        v2f a, b;
        a.x = A[m * lda + ka];
        a.y = A[m * lda + ka + 1];
        b.x = B[(kbase + ka) * 64 + n];
        b.y = B[(kbase + ka + 1) * 64 + n];
        acc = wmma_f32_4(a, b, acc);
    }
    return acc;
}

// -------------------------- degree / normalization -------------------------

__global__ void deg_init_kernel(float* __restrict__ deg) {
    int i = blockIdx.x * blockDim.x + threadIdx.x;
    if (i < NN) deg[i] = 1.0f;   // self-loop weight 1
}

__global__ void deg_accum_kernel(const float* __restrict__ w,
                                 const int* __restrict__ ei,
                                 float* __restrict__ deg) {
    int e = blockIdx.x * blockDim.x + threadIdx.x;
    if (e < EE) {
        int col = ei[EE + e];
        unsafeAtomicAdd(&deg[col], w[e]);
    }
}

__global__ void dinv_kernel(float* __restrict__ deg) {
    int i = blockIdx.x * blockDim.x + threadIdx.x;
    if (i < NN) {
        float d = deg[i];
        deg[i] = (d > 0.0f) ? rsqrtf(d) : 0.0f;   // in-place deg -> deg^-1/2
    }
}

// ------------------------- generic 16x16 WMMA GEMM -------------------------
// D[m, n] = sum_k A[m,k] * B[k,n] (+ bias[n]); one wave per 16x16 tile.
// C/D layout: VGPR j = row M=j (lanes 0-15) / M=j+8 (lanes 16-31), N = lane&15.

__global__ void gemm16_kernel(const float* __restrict__ A, int lda,
                              const float* __restrict__ B,    // ldb = 64
                              const float* __restrict__ bias, // may be null
                              float* __restrict__ D,          // ldd = 64
                              int K) {
    const int lane    = threadIdx.x;       // blockDim.x == 32 (one wave)
    const int half    = lane >> 4;
    const int l16     = lane & 15;
    const int rowBase = blockIdx.x * 16;
    const int colBase = blockIdx.y * 16;
    const int n       = colBase + l16;
    const int m       = rowBase + l16;

    v8f acc;
    float c0 = bias ? bias[n] : 0.0f;
#pragma unroll
    for (int j = 0; j < 8; ++j) acc[j] = c0;

    acc = gemm_seg(acc, A, lda, m, B, n, 0, K, half);

#pragma unroll
    for (int j = 0; j < 8; ++j) {
        int mr = rowBase + j + 8 * half;
        D[mr * 64 + n] = acc[j];
    }
}

// Hidden-state GEMM on the virtual concat [epi | info | hidden | x] (K = 200),
// epilogue: ReLU.  Four branch-free contiguous segment loops (boundaries are
// compile-time constants, each a multiple of 4).
__global__ void gemm_concat_relu_kernel(const float* __restrict__ epi,
                                        const float* __restrict__ info,
                                        const float* __restrict__ hid,
                                        const float* __restrict__ x,
                                        const float* __restrict__ B,    // hid_w [200,64]
                                        const float* __restrict__ bias, // hid_b [64]
                                        float* __restrict__ D) {        // z [N,64]
    const int lane    = threadIdx.x;
    const int half    = lane >> 4;
    const int l16     = lane & 15;
    const int rowBase = blockIdx.x * 16;
    const int colBase = blockIdx.y * 16;
    const int n       = colBase + l16;
    const int m       = rowBase + l16;

    v8f acc;
    float c0 = bias[n];
#pragma unroll
    for (int j = 0; j < 8; ++j) acc[j] = c0;

    acc = gemm_seg(acc, epi,  HH,  m, B, n,   0, HH,  half);
    acc = gemm_seg(acc, info, HH,  m, B, n,  64, HH,  half);
    acc = gemm_seg(acc, hid,  HH,  m, B, n, 128, HH,  half);
    acc = gemm_seg(acc, x,    FIN, m, B, n, 192, FIN, half);

#pragma unroll
    for (int j = 0; j < 8; ++j) {
        int mr = rowBase + j + 8 * half;
        D[mr * 64 + n] = fmaxf(acc[j], 0.0f);
    }
}

// ------------------------------- GCN pieces --------------------------------

// epi = gcn_b + dinv^2 * h  (bias + self-loop term), then edges scattered in.
__global__ void epi_init_kernel(const float* __restrict__ h,
                                const float* __restrict__ dinv,
                                const float* __restrict__ gcn_b,
                                float* __restrict__ epi) {
    int t = blockIdx.x * blockDim.x + threadIdx.x;
    if (t < NN * HH) {
        int i = t >> 6, f = t & 63;
        float di = dinv[i];
        epi[t] = gcn_b[f] + di * di * h[t];
    }
}

__global__ void gcn_scatter_kernel(const float* __restrict__ h,
                                   const float* __restrict__ dinv,
                                   const float* __restrict__ w,
                                   const int* __restrict__ ei,
                                   float* __restrict__ epi) {
    int t = blockIdx.x * blockDim.x + threadIdx.x;
    if (t < EE * HH) {
        int e = t >> 6, f = t & 63;
        int row = ei[e], col = ei[EE + e];
        float nrm = dinv[row] * w[e] * dinv[col];
        unsafeAtomicAdd(&epi[col * 64 + f], nrm * h[row * 64 + f]);
    }
}

// ------------------------------- NNConv edges ------------------------------
// 4 edges per 256-thread block; per-edge MLP weights staged in LDS; the
// per-edge theta matrix is generated and consumed in registers (never stored).

__global__ void nnconv1_edge_kernel(const float* __restrict__ xm,   // [N,8]
                                    const float* __restrict__ attr, // [E,2]
                                    const int* __restrict__ eim,
                                    const float* __restrict__ W1,   // [2,512]
                                    const float* __restrict__ b1,   // [512]
                                    float* __restrict__ h1) {
    __shared__ float sW0[512], sW1[512], sB[512], sX[4 * FIN];
    for (int i = threadIdx.x; i < 512; i += 256) {
        sW0[i] = W1[i];
        sW1[i] = W1[512 + i];
        sB[i]  = b1[i];
    }
    int le = threadIdx.x >> 6;     // local edge 0..3
    int o  = threadIdx.x & 63;     // output feature
    int e  = blockIdx.x * 4 + le;
    bool valid = (e < EE);
    int src = 0, dst = 0;
    float a0 = 0.f, a1 = 0.f;
    if (valid) {
        src = eim[e]; dst = eim[EE + e];
        a0 = attr[2 * e]; a1 = attr[2 * e + 1];
        if (o < FIN) sX[le * FIN + o] = xm[src * FIN + o];
    }
    __syncthreads();
    if (valid) {
        float acc = 0.0f;
#pragma unroll
        for (int i = 0; i < FIN; ++i) {
            float th = fmaf(a0, sW0[i * 64 + o], fmaf(a1, sW1[i * 64 + o], sB[i * 64 + o]));
            th = fmaxf(th, 0.0f);
            acc = fmaf(sX[le * FIN + i], th, acc);
        }
        unsafeAtomicAdd(&h1[dst * 64 + o], acc);
    }
}

__global__ void nnconv2_edge_kernel(const float* __restrict__ h1,   // [N,64]
                                    const float* __restrict__ attr, // [E,2]
                                    const int* __restrict__ eim,
                                    const float* __restrict__ W2,   // [2,4096]
                                    const float* __restrict__ b2,   // [4096]
                                    float* __restrict__ info) {
    __shared__ float sW0[4096], sW1[4096], sB[4096], sH[4 * 64];
    for (int i = threadIdx.x; i < 4096; i += 256) {
        sW0[i] = W2[i];
        sW1[i] = W2[4096 + i];
        sB[i]  = b2[i];
    }
    int le = threadIdx.x >> 6;
    int o  = threadIdx.x & 63;
    int e  = blockIdx.x * 4 + le;
    bool valid = (e < EE);
    int src = 0, dst = 0;
    float a0 = 0.f, a1 = 0.f;
    if (valid) {
        src = eim[e]; dst = eim[EE + e];
        a0 = attr[2 * e]; a1 = attr[2 * e + 1];
        sH[le * 64 + o] = h1[src * 64 + o];
    }
    __syncthreads();
    if (valid) {
        float acc = 0.0f;
        const float* hrow = &sH[le * 64];
#pragma unroll 8
        for (int i = 0; i < 64; ++i) {
            float th = fmaf(a0, sW0[i * 64 + o], fmaf(a1, sW1[i * 64 + o], sB[i * 64 + o]));
            th = fmaxf(th, 0.0f);
            acc = fmaf(hrow[i], th, acc);
        }
        unsafeAtomicAdd(&info[dst * 64 + o], acc);
    }
}

// ------------------------- L2-normalize + score head -----------------------
// One wave32 per node; wave reduction via shfl_xor (width 32 on gfx1250).

__global__ void finalize_kernel(const float* __restrict__ z,
                                const float* __restrict__ hidden_prev,
                                const float* __restrict__ x,
                                const float* __restrict__ score_w, // [136]
                                const float* __restrict__ score_b, // [1]
                                float* __restrict__ scores,        // [N]
                                float* __restrict__ hidden_new) {  // [N,64]
    int r = blockIdx.x;
    int l = threadIdx.x;           // 0..31
    float z0 = z[r * 64 + l];
    float z1 = z[r * 64 + 32 + l];
    float ss = z0 * z0 + z1 * z1;
#pragma unroll
    for (int m = 16; m >= 1; m >>= 1) ss += __shfl_xor(ss, m, 32);
    float inv = 1.0f / fmaxf(sqrtf(ss), 1e-12f);
    float hn0 = z0 * inv, hn1 = z1 * inv;
    hidden_new[r * 64 + l]      = hn0;
    hidden_new[r * 64 + 32 + l] = hn1;

    float dot = hidden_prev[r * 64 + l]      * score_w[l]
              + hidden_prev[r * 64 + 32 + l] * score_w[32 + l]
              + hn0 * score_w[64 + l]
              + hn1 * score_w[96 + l];
    if (l < FIN) dot = fmaf(x[r * FIN + l], score_w[128 + l], dot);
#pragma unroll
    for (int m = 16; m >= 1; m >>= 1) dot += __shfl_xor(dot, m, 32);
    if (l == 0) scores[r] = fmaxf(dot + score_b[0], 0.0f);
}

// --------------------------------- launch ----------------------------------

extern "C" void kernel_launch(void* const* d_in, const int* in_sizes, int n_in,
                              void* d_out, int out_size, void* d_ws, size_t ws_size,
                              hipStream_t stream) {
    (void)in_sizes; (void)n_in; (void)out_size; (void)ws_size;

    const float* x        = (const float*)d_in[0];
    const float* ew       = (const float*)d_in[1];
    const float* xm       = (const float*)d_in[2];
    const float* attr     = (const float*)d_in[3];
    const float* hidden   = (const float*)d_in[4];
    const float* gcn_w    = (const float*)d_in[5];
    const float* gcn_b    = (const float*)d_in[6];
    const float* nn1_w    = (const float*)d_in[7];
    const float* nn1_b    = (const float*)d_in[8];
    const float* root1    = (const float*)d_in[9];
    const float* bias1    = (const float*)d_in[10];
    const float* nn2_w    = (const float*)d_in[11];
    const float* nn2_b    = (const float*)d_in[12];
    const float* root2    = (const float*)d_in[13];
    const float* bias2    = (const float*)d_in[14];
    const float* hid_w    = (const float*)d_in[15];
    const float* hid_b    = (const float*)d_in[16];
    const float* score_w  = (const float*)d_in[17];
    const float* score_b  = (const float*)d_in[18];
    const int*   ei       = (const int*)d_in[19];
    const int*   eim      = (const int*)d_in[20];

    float* scores     = (float*)d_out;        // [N]
    float* hidden_new = (float*)d_out + NN;   // [N,64]

    // workspace layout (floats)
    float* ws   = (float*)d_ws;
    float* dinv = ws;                          // [N]      (deg, then deg^-1/2)
    float* h    = ws + 10240;                  // [N,64]   x @ gcn_w
    float* epi  = h   + NN * HH;               // [N,64]
    float* h1   = epi + NN * HH;               // [N,64]
    float* info = h1  + NN * HH;               // [N,64]
    float* z    = info + NN * HH;              // [N,64]

    const dim3 wave(32, 1, 1);
    const dim3 tilesN(NN / 16, HH / 16, 1);    // 625 x 4 16x16 tiles

    // ---- GCN ----
    deg_init_kernel<<<(NN + 255) / 256, 256, 0, stream>>>(dinv);
    deg_accum_kernel<<<(EE + 255) / 256, 256, 0, stream>>>(ew, ei, dinv);
    dinv_kernel<<<(NN + 255) / 256, 256, 0, stream>>>(dinv);
    gemm16_kernel<<<tilesN, wave, 0, stream>>>(x, FIN, gcn_w, nullptr, h, FIN);
    epi_init_kernel<<<(NN * HH + 255) / 256, 256, 0, stream>>>(h, dinv, gcn_b, epi);
    gcn_scatter_kernel<<<(EE * HH + 255) / 256, 256, 0, stream>>>(h, dinv, ew, ei, epi);

    // ---- NNConv(8 -> 64): h1 = agg + xm @ root1 + bias1 ----
    gemm16_kernel<<<tilesN, wave, 0, stream>>>(xm, FIN, root1, bias1, h1, FIN);
    nnconv1_edge_kernel<<<(EE + 3) / 4, 256, 0, stream>>>(xm, attr, eim, nn1_w, nn1_b, h1);

    // ---- NNConv(64 -> 64): info = agg + h1 @ root2 + bias2 ----
    gemm16_kernel<<<tilesN, wave, 0, stream>>>(h1, HH, root2, bias2, info, HH);
    nnconv2_edge_kernel<<<(EE + 3) / 4, 256, 0, stream>>>(h1, attr, eim, nn2_w, nn2_b, info);

    // ---- hidden state: z = relu([epi|info|hidden|x] @ hid_w + hid_b) ----
    gemm_concat_relu_kernel<<<tilesN, wave, 0, stream>>>(epi, info, hidden, x,
                                                         hid_w, hid_b, z);

    // ---- L2 normalize + score head ----
    finalize_kernel<<<NN, 32, 0, stream>>>(z, hidden, x, score_w, score_b,
                                           scores, hidden_new);
}